// Gait_Graph2_Block_6150393168643
// MI455X (gfx1250) — compile-verified
//
#include <hip/hip_runtime.h>
#include <hip/hip_bf16.h>

// ---------------------------------------------------------------------------
// Problem constants (match reference)
// ---------------------------------------------------------------------------
#define BB     32
#define CC     64
#define TT     1024
#define HH     2
#define DD     64
#define DIM2   128
#define KK     9
#define NN     (BB * TT)          // 32768 nodes
#define EE     524288             // edges
#define EN     (EE + NN)          // edges + self loops = 557056
#define XN_SZ  (BB * CC * TT)     // 2,097,152
#define NG     (NN * DIM2)        // 4,194,304
#define KFLAT  (CC * KK)          // 576
#define EPSV   1e-5f

typedef float v8f  __attribute__((ext_vector_type(8)));
typedef float v2f  __attribute__((ext_vector_type(2)));

// fp32 WMMA: D(16x16) = A(16x4) * B(4x16) + C   -- exact fp32 semantics
static __device__ __forceinline__ v8f wmma4(v2f a, v2f b, v8f c) {
  return __builtin_amdgcn_wmma_f32_16x16x4_f32(
      /*neg_a=*/false, a, /*neg_b=*/false, b,
      /*c_mod=*/(short)0, c, /*reuse_a=*/false, /*reuse_b=*/false);
}

// ---------------------------------------------------------------------------
// 1) bn0 stats: per-channel mean/var over (B,T); emit scale/shift
// ---------------------------------------------------------------------------
__global__ __launch_bounds__(256)
void k_bn0_stats(const float* __restrict__ x, const float* __restrict__ g,
                 const float* __restrict__ be, float* __restrict__ sc,
                 float* __restrict__ sh) {
  __shared__ float s1[256], s2[256];
  const int c = blockIdx.x;                 // 64 channels
  float a = 0.f, q = 0.f;
  for (int j = threadIdx.x; j < BB * TT; j += 256) {
    const int b = j >> 10, t = j & 1023;
    const float v = x[b * (CC * TT) + c * TT + t];
    a += v; q += v * v;
  }
  s1[threadIdx.x] = a; s2[threadIdx.x] = q;
  __syncthreads();
  for (int s = 128; s > 0; s >>= 1) {
    if (threadIdx.x < s) { s1[threadIdx.x] += s1[threadIdx.x + s];
                           s2[threadIdx.x] += s2[threadIdx.x + s]; }
    __syncthreads();
  }
  if (threadIdx.x == 0) {
    const float inv = 1.f / (float)(BB * TT);
    const float mu = s1[0] * inv;
    const float var = s2[0] * inv - mu * mu;
    const float a0 = g[c] * rsqrtf(var + EPSV);
    sc[c] = a0;
    sh[c] = be[c] - mu * a0;
  }
}

__global__ __launch_bounds__(256)
void k_bn0_apply(const float* __restrict__ x, const float* __restrict__ sc,
                 const float* __restrict__ sh, float* __restrict__ xn) {
  const int i = blockIdx.x * 256 + threadIdx.x;
  if (i >= XN_SZ) return;
  const int c = (i >> 10) & (CC - 1);       // [B,C,T] flat: channel bits
  xn[i] = x[i] * sc[c] + sh[c];
}

// ---------------------------------------------------------------------------
// 2) conv1d SAME (Cin=64, K=9, Cout=128) + ReLU via implicit-GEMM WMMA fp32
//    per-batch GEMM: out[o,t] = W[o, c*9+k] * P[c*9+k, t];  K = 576
// ---------------------------------------------------------------------------
__global__ __launch_bounds__(256)
void k_conv_wmma(const float* __restrict__ xn, const float* __restrict__ w,
                 const float* __restrict__ bias, float* __restrict__ res) {
  const int lane = threadIdx.x & 31;
  const int wid  = threadIdx.x >> 5;
  const int tile = blockIdx.x * 8 + wid;    // 32 * 8 * 64 = 16384 tiles
  const int b  = tile >> 9;                 // /(8*64)
  const int r0 = tile & 511;
  const int ot = r0 >> 6;                   // o-tile 0..7
  const int tt = r0 & 63;                   // t-tile 0..63
  const int o0 = ot * 16, t0 = tt * 16;
  const int half = lane >> 4, l15 = lane & 15;

  v8f acc = {};
  const float* xb = xn + b * (CC * TT);
  for (int kk = 0; kk < KFLAT; kk += 4) {
    // A tile: W[o0+l15][kk + half*2 + r], r=0,1 (contiguous -> float2)
    const v2f a = *(const v2f*)&w[(o0 + l15) * KFLAT + kk + half * 2];
    // B tile: P[kk + half*2 + r][t0 + l15]  (im2col with zero pad)
    v2f bvec;
    #pragma unroll
    for (int r = 0; r < 2; ++r) {
      const int ck = kk + half * 2 + r;
      const int c  = ck / KK;
      const int k  = ck - c * KK;
      const int ts = t0 + l15 + k - (KK / 2);
      bvec[r] = (ts >= 0 && ts < TT) ? xb[c * TT + ts] : 0.f;
    }
    acc = wmma4(a, bvec, acc);
  }
  #pragma unroll
  for (int r = 0; r < 8; ++r) {
    const int o = o0 + r + half * 8;
    const float v = acc[r] + bias[o];
    res[b * (DIM2 * TT) + o * TT + t0 + l15] = v > 0.f ? v : 0.f;
  }
}

// ---------------------------------------------------------------------------
// 3) generic GEMM: out[M,N] = A[M,K] @ W[N,K]^T (+bias), fp32 WMMA
//    wave per 16x16 tile; M=32768, N=128, K in {64,128}
// ---------------------------------------------------------------------------
__global__ __launch_bounds__(256)
void k_gemm_nt_wmma(const float* __restrict__ A, const float* __restrict__ W,
                    const float* __restrict__ bias, float* __restrict__ out,
                    int K) {
  const int lane = threadIdx.x & 31;
  const int wid  = threadIdx.x >> 5;
  const int tile = blockIdx.x * 8 + wid;    // (32768/16)*(128/16) = 16384
  const int mt = tile >> 3;                 // 0..2047
  const int nt = tile & 7;                  // 0..7
  const int m0 = mt * 16, n0 = nt * 16;
  const int half = lane >> 4, l15 = lane & 15;

  v8f acc = {};
  const float* arow = A + (m0 + l15) * K;
  const float* wrow = W + (n0 + l15) * K;
  for (int kk = 0; kk < K; kk += 4) {
    const v2f a = *(const v2f*)&arow[kk + half * 2];  // A[m][k..k+1]
    const v2f b = *(const v2f*)&wrow[kk + half * 2];  // B[k][n] = W[n][k]
    acc = wmma4(a, b, acc);
  }
  const float bv = bias ? bias[n0 + l15] : 0.f;
  #pragma unroll
  for (int r = 0; r < 8; ++r) {
    out[(m0 + r + half * 8) * DIM2 + n0 + l15] = acc[r] + bv;
  }
}

// ---------------------------------------------------------------------------
// 4) GAT: init output with bias, zero denominators
// ---------------------------------------------------------------------------
__global__ __launch_bounds__(256)
void k_gat_init(float* __restrict__ gout, const float* __restrict__ bias,
                float* __restrict__ den) {
  const int i = blockIdx.x * 256 + threadIdx.x;
  if (i < NG)                 gout[i] = bias[i & (DIM2 - 1)];
  else if (i < NG + NN * HH)  den[i - NG] = 0.f;
}

// ---------------------------------------------------------------------------
// 5) GAT edge pass 1: logits -> p = exp(logit); denom[dst,h] += p
//    wave32 per edge; lanes read coalesced float4; heads = lane halves
// ---------------------------------------------------------------------------
__global__ __launch_bounds__(256)
void k_edge_logits(const int* __restrict__ ei, const float* __restrict__ xl,
                   const float* __restrict__ xr, const float* __restrict__ att,
                   float* __restrict__ pbuf, float* __restrict__ den) {
  const int lane = threadIdx.x & 31;
  const int e = blockIdx.x * 8 + (threadIdx.x >> 5);
  if (e >= EN) return;
  int src, dst;
  if (e < EE) { src = ei[e]; dst = ei[EE + e]; }
  else        { src = e - EE; dst = src; }

  const float4 vl = *(const float4*)&xl[src * DIM2 + lane * 4];
  const float4 vr = *(const float4*)&xr[dst * DIM2 + lane * 4];
  const float4 at = *(const float4*)&att[lane * 4];
  float s = 0.f;
  {
    float m;
    m = vl.x + vr.x; s += (m > 0.f ? m : 0.2f * m) * at.x;
    m = vl.y + vr.y; s += (m > 0.f ? m : 0.2f * m) * at.y;
    m = vl.z + vr.z; s += (m > 0.f ? m : 0.2f * m) * at.z;
    m = vl.w + vr.w; s += (m > 0.f ? m : 0.2f * m) * at.w;
  }
  // reduce within each 16-lane half (one head per half)
  s += __shfl_xor(s, 1);
  s += __shfl_xor(s, 2);
  s += __shfl_xor(s, 4);
  s += __shfl_xor(s, 8);
  if ((lane & 15) == 0) {
    const int h = lane >> 4;
    const float p = __expf(s);            // softmax shift-invariant: no max
    pbuf[e * HH + h] = p;
    atomicAdd(&den[dst * HH + h], p);
  }
}

// ---------------------------------------------------------------------------
// 6) GAT edge pass 2: gout[dst] += (p/denom[dst]) * xl[src]
// ---------------------------------------------------------------------------
__global__ __launch_bounds__(256)
void k_edge_aggr(const int* __restrict__ ei, const float* __restrict__ xl,
                 const float* __restrict__ pbuf, const float* __restrict__ den,
                 float* __restrict__ gout) {
  const int lane = threadIdx.x & 31;
  const int e = blockIdx.x * 8 + (threadIdx.x >> 5);
  if (e >= EN) return;
  int src, dst;
  if (e < EE) { src = ei[e]; dst = ei[EE + e]; }
  else        { src = e - EE; dst = src; }

  const int h = lane >> 4;
  const float alpha = pbuf[e * HH + h] / den[dst * HH + h];
  const float4 v = *(const float4*)&xl[src * DIM2 + lane * 4];
  float* o = &gout[dst * DIM2 + lane * 4];
  atomicAdd(o + 0, alpha * v.x);
  atomicAdd(o + 1, alpha * v.y);
  atomicAdd(o + 2, alpha * v.z);
  atomicAdd(o + 3, alpha * v.w);
}

// ---------------------------------------------------------------------------
// 7) column batch-norm stats on [rows, 128]; emit scale/shift per column
// ---------------------------------------------------------------------------
__global__ __launch_bounds__(256)
void k_bncol_stats(const float* __restrict__ in, const float* __restrict__ g,
                   const float* __restrict__ be, float* __restrict__ a,
                   float* __restrict__ b2) {
  __shared__ float s1[256], s2[256];
  const int c = blockIdx.x;                 // 128 columns
  float s = 0.f, q = 0.f;
  for (int r = threadIdx.x; r < NN; r += 256) {
    const float v = in[r * DIM2 + c];
    s += v; q += v * v;
  }
  s1[threadIdx.x] = s; s2[threadIdx.x] = q;
  __syncthreads();
  for (int st = 128; st > 0; st >>= 1) {
    if (threadIdx.x < st) { s1[threadIdx.x] += s1[threadIdx.x + st];
                            s2[threadIdx.x] += s2[threadIdx.x + st]; }
    __syncthreads();
  }
  if (threadIdx.x == 0) {
    const float inv = 1.f / (float)NN;
    const float mu = s1[0] * inv;
    const float var = s2[0] * inv - mu * mu;
    const float sc = g[c] * rsqrtf(var + EPSV);
    a[c]  = sc;
    b2[c] = be[c] - mu * sc;
  }
}

// 8) tmp = residual + relu(bn(gout))       (all flat; channel = i & 127)
__global__ __launch_bounds__(256)
void k_add_res_bn(const float* __restrict__ res, const float* __restrict__ v,
                  const float* __restrict__ a, const float* __restrict__ b,
                  float* __restrict__ out) {
  const int i = blockIdx.x * 256 + threadIdx.x;
  if (i >= NG) return;
  const int c = i & (DIM2 - 1);
  const float h = v[i] * a[c] + b[c];
  out[i] = res[i] + (h > 0.f ? h : 0.f);
}

// ---------------------------------------------------------------------------
extern "C" void kernel_launch(void* const* d_in, const int* in_sizes, int n_in,
                              void* d_out, int out_size, void* d_ws, size_t ws_size,
                              hipStream_t stream) {
  const float* x        = (const float*)d_in[0];
  const int*   ei       = (const int*)  d_in[1];
  const float* bn0_g    = (const float*)d_in[2];
  const float* bn0_b    = (const float*)d_in[3];
  const float* conv_w   = (const float*)d_in[4];
  const float* conv_b   = (const float*)d_in[5];
  const float* gat_wl   = (const float*)d_in[6];
  const float* gat_wr   = (const float*)d_in[7];
  const float* gat_att  = (const float*)d_in[8];
  const float* gat_bias = (const float*)d_in[9];
  const float* bn1_g    = (const float*)d_in[10];
  const float* bn1_b    = (const float*)d_in[11];
  const float* cheb_w   = (const float*)d_in[12];
  const float* cheb_b   = (const float*)d_in[13];
  float* out = (float*)d_out;

  // workspace layout (floats)
  float* ws   = (float*)d_ws;
  float* xn   = ws;                          // 2,097,152
  float* xl   = xn  + XN_SZ;                 // 4,194,304
  float* xr   = xl  + NG;                    // 4,194,304
  float* res  = xr  + NG;                    // 4,194,304
  float* gout = res + NG;                    // 4,194,304
  float* pbuf = gout + NG;                   // 1,114,112
  float* den  = pbuf + (size_t)EN * HH;      // 65,536
  float* st   = den + NN * HH;               // stats: 1024
  float* sc0 = st, *sh0 = st + 64;
  float* a1  = st + 128, *b1 = st + 256;
  float* a2  = st + 384, *b2 = st + 512;
  float* tmp  = xl;                          // reuse after edge passes
  float* out2 = xr;                          // reuse after edge passes

  // 1. bn0
  k_bn0_stats<<<CC, 256, 0, stream>>>(x, bn0_g, bn0_b, sc0, sh0);
  k_bn0_apply<<<XN_SZ / 256, 256, 0, stream>>>(x, sc0, sh0, xn);
  // 2. residual = relu(conv1d(xn))  [WMMA implicit GEMM]
  k_conv_wmma<<<2048, 256, 0, stream>>>(xn, conv_w, conv_b, res);
  // 3. xl/xr GEMMs [WMMA]
  k_gemm_nt_wmma<<<2048, 256, 0, stream>>>(xn, gat_wl, nullptr, xl, CC);
  k_gemm_nt_wmma<<<2048, 256, 0, stream>>>(xn, gat_wr, nullptr, xr, CC);
  // 4. GAT softmax-aggregate over edges
  k_gat_init<<<(NG + NN * HH + 255) / 256, 256, 0, stream>>>(gout, gat_bias, den);
  k_edge_logits<<<EN / 8, 256, 0, stream>>>(ei, xl, xr, gat_att, pbuf, den);
  k_edge_aggr  <<<EN / 8, 256, 0, stream>>>(ei, xl, pbuf, den, gout);
  // 5. bn1 + relu, add residual (flat)
  k_bncol_stats<<<DIM2, 256, 0, stream>>>(gout, bn1_g, bn1_b, a1, b1);
  k_add_res_bn<<<NG / 256, 256, 0, stream>>>(res, gout, a1, b1, tmp);
  // 6. cheb linear [WMMA] + bn1 + relu + residual -> d_out
  k_gemm_nt_wmma<<<2048, 256, 0, stream>>>(tmp, cheb_w, cheb_b, out2, DIM2);
  k_bncol_stats<<<DIM2, 256, 0, stream>>>(out2, bn1_g, bn1_b, a2, b2);
  k_add_res_bn<<<NG / 256, 256, 0, stream>>>(res, out2, a2, b2, out);
}